// L2Baseline_44693429682542
// MI455X (gfx1250) — compile-verified
//
#include <hip/hip_runtime.h>
#include <hip/hip_bf16.h>

typedef __attribute__((ext_vector_type(2))) float v2f;
typedef __attribute__((ext_vector_type(8))) float v8f;

// Problem constants (from setup_inputs: [64, 2048, 512] f32)
#define BATCH 64
#define SEQ   2048
#define DIM   512

// Pass 1: per-(b, s) squared L2 distance of consecutive rows, via f32 WMMA
// reduction (D = A x ones accumulates row sums of A in full f32 precision).
// One wave32 computes 16 consecutive timesteps of one batch.
__global__ __launch_bounds__(128) void l2d_pass1(const float* __restrict__ x,
                                                 float* __restrict__ out) {
    const int lane   = threadIdx.x & 31;
    const int waveIn = threadIdx.x >> 5;                 // 4 waves / block
    const int wave   = blockIdx.x * 4 + waveIn;          // global wave id
    const int tilesPerBatch = SEQ / 16;                  // 128
    const int b  = wave / tilesPerBatch;
    const int s0 = (wave % tilesPerBatch) * 16;

    const int m = lane & 15;          // timestep within tile (A-matrix M)
    const int g = lane >> 4;          // 0: dims d0..d0+3, 1: dims d0+4..d0+7

    const int s  = s0 + m;
    const int sp = (s == 0) ? 0 : (s - 1);               // clamp -> d[0] = 0

    const float* __restrict__ cur  = x + ((size_t)b * SEQ + s)  * DIM;
    const float* __restrict__ prev = x + ((size_t)b * SEQ + sp) * DIM;

    v8f acc0 = {};
    v8f acc1 = {};
    const v2f ones = {1.0f, 1.0f};

    // 8 dims per iteration: one float4 load per row, two WMMAs.
    #pragma unroll 4
    for (int d0 = 0; d0 < DIM; d0 += 8) {
        const float4 c4 = *reinterpret_cast<const float4*>(cur  + d0 + 4 * g);
        const float4 p4 = *reinterpret_cast<const float4*>(prev + d0 + 4 * g);
        const float dx0 = c4.x - p4.x;
        const float dx1 = c4.y - p4.y;
        const float dx2 = c4.z - p4.z;
        const float dx3 = c4.w - p4.w;
        v2f a0 = { dx0 * dx0, dx1 * dx1 };
        v2f a1 = { dx2 * dx2, dx3 * dx3 };
        // D[m][n] = sum_k A[m][k] * 1 + C  -> per-row partial sums, f32 RNE.
        acc0 = __builtin_amdgcn_wmma_f32_16x16x4_f32(
            false, a0, false, ones, (short)0, acc0, false, false);
        acc1 = __builtin_amdgcn_wmma_f32_16x16x4_f32(
            false, a1, false, ones, (short)0, acc1, false, false);
    }

    const v8f accf = acc0 + acc1;

    // D layout (16x16 f32 C/D): VGPR j, lanes 0-15 -> M=j (N=lane),
    // lanes 16-31 -> M=j+8. All N columns identical (B is all ones), so
    // lane 0 holds d[s0..s0+7], lane 16 holds d[s0+8..s0+15].
    float* __restrict__ op = out + (size_t)b * SEQ + s0;
    if (lane == 0) {
        #pragma unroll
        for (int j = 0; j < 8; ++j) op[j] = accf[j];
    } else if (lane == 16) {
        #pragma unroll
        for (int j = 0; j < 8; ++j) op[8 + j] = accf[j];
    }
}

// Pass 2: per-batch max over SEQ values, then normalize in place.
__global__ __launch_bounds__(256) void l2d_pass2(float* __restrict__ d) {
    __shared__ float red[256];
    const int b = blockIdx.x;
    const int t = threadIdx.x;
    float* __restrict__ row = d + (size_t)b * SEQ;

    float v[SEQ / 256];
    float mx = 0.0f;   // distances are >= 0 and d[0] == 0
    #pragma unroll
    for (int i = 0; i < SEQ / 256; ++i) {
        v[i] = row[t + i * 256];
        mx = fmaxf(mx, v[i]);
    }
    red[t] = mx;
    __syncthreads();
    #pragma unroll
    for (int off = 128; off > 0; off >>= 1) {
        if (t < off) red[t] = fmaxf(red[t], red[t + off]);
        __syncthreads();
    }
    const float mxb = red[0];
    #pragma unroll
    for (int i = 0; i < SEQ / 256; ++i) {
        row[t + i * 256] = v[i] / mxb;   // match reference divide semantics
    }
}

extern "C" void kernel_launch(void* const* d_in, const int* in_sizes, int n_in,
                              void* d_out, int out_size, void* d_ws, size_t ws_size,
                              hipStream_t stream) {
    (void)in_sizes; (void)n_in; (void)out_size; (void)d_ws; (void)ws_size;
    const float* x = (const float*)d_in[0];
    float* out = (float*)d_out;

    // Pass 1: BATCH * (SEQ/16) waves = 8192 waves, 4 waves (128 thr) / block.
    const int totalWaves = BATCH * (SEQ / 16);
    l2d_pass1<<<totalWaves / 4, 128, 0, stream>>>(x, out);

    // Pass 2: one block per batch.
    l2d_pass2<<<BATCH, 256, 0, stream>>>(out);
}